// SimpleGINlayer_87222195848273
// MI455X (gfx1250) — compile-verified
//
#include <hip/hip_runtime.h>
#include <stdint.h>

#define HIDDEN 64
#define THREADS 256
#define WAVES_PER_BLOCK (THREADS / 32)
#define CHUNK 32   // edges staged per wave per buffer

// ---------------------------------------------------------------------------
// Hardware FP32 atomic add (no-return form -> STOREcnt, no CAS loop).
// ---------------------------------------------------------------------------
__device__ __forceinline__ void hw_atomic_add_f32(float* p, float v) {
    asm volatile("global_atomic_add_f32 %0, %1, off" :: "v"(p), "v"(v) : "memory");
}

// ---------------------------------------------------------------------------
// Kernel 1: out = (1 + eps) * x   (vectorized float4, fully writes d_out)
// ---------------------------------------------------------------------------
__global__ __launch_bounds__(THREADS)
void gin_combine(const float* __restrict__ x,
                 const float* __restrict__ eps,
                 float* __restrict__ out,
                 int total4) {
    const float s = 1.0f + eps[0];
    int i = blockIdx.x * THREADS + threadIdx.x;
    if (i < total4) {
        float4 v = ((const float4*)x)[i];
        float4 r;
        r.x = s * v.x; r.y = s * v.y; r.z = s * v.z; r.w = s * v.w;
        ((float4*)out)[i] = r;
    }
}

// ---------------------------------------------------------------------------
// Kernel 2: scatter-add of x[col] into out[row].
// Wave-private double-buffered async staging of edge indices into LDS:
//   global_load_async_to_lds_b64  (ASYNCcnt)  +  s_wait_asynccnt
// 16 lanes per edge, each lane carries a float4 (16 * 4 = 64 floats/row).
// ---------------------------------------------------------------------------
__global__ __launch_bounds__(THREADS)
void gin_scatter(const float* __restrict__ x,
                 const long long* __restrict__ rows,
                 const long long* __restrict__ cols,
                 float* __restrict__ out,
                 long long E,
                 int totalWaves) {
    // [wave][buffer][edge][{row,col}] as u64
    __shared__ unsigned long long smem[WAVES_PER_BLOCK * 2 * CHUNK * 2];

    const int lane  = threadIdx.x & 31;
    const int wib   = threadIdx.x >> 5;                 // wave in block
    const int gw    = blockIdx.x * WAVES_PER_BLOCK + wib;
    const long long nChunks = (E + CHUNK - 1) / CHUNK;

    unsigned long long* mybuf = &smem[wib * (2 * CHUNK * 2)];
    // Generic LDS pointer: low 32 bits are the LDS byte address (ISA 10.2).
    const uint32_t ldsBase = (uint32_t)(uintptr_t)mybuf;

    // Stage chunk c's 32 (row,col) pairs into buffer b, one edge per lane.
    auto stage = [&](long long c, int b) {
        long long e = c * CHUNK + lane;
        if (e >= E) e = E - 1;                          // clamp: safe duplicate
        const long long* ra = rows + e;
        const long long* ca = cols + e;
        uint32_t ldsR = ldsBase + (uint32_t)(b * CHUNK * 2 + lane * 2) * 8u;
        uint32_t ldsC = ldsR + 8u;
        asm volatile("global_load_async_to_lds_b64 %0, %1, off"
                     :: "v"(ldsR), "v"(ra) : "memory");
        asm volatile("global_load_async_to_lds_b64 %0, %1, off"
                     :: "v"(ldsC), "v"(ca) : "memory");
    };

    const int g  = lane >> 4;       // which of the 2 edges this half-wave owns
    const int l4 = lane & 15;       // lane within the 16-lane edge group

    long long c = gw;
    if (c >= nChunks) return;
    stage(c, 0);
    int buf = 0;

    for (; c < nChunks; c += totalWaves) {
        const long long cn = c + totalWaves;
        if (cn < nChunks) {
            stage(cn, buf ^ 1);     // prefetch next chunk into other buffer
            // 2 ops outstanding for next chunk; wait until current chunk done.
            asm volatile("s_wait_asynccnt 0x2" ::: "memory");
        } else {
            asm volatile("s_wait_asynccnt 0x0" ::: "memory");
        }

        const unsigned long long* bb = mybuf + buf * CHUNK * 2;

        #pragma unroll 4
        for (int i = 0; i < CHUNK / 2; ++i) {
            const int el = i * 2 + g;
            const long long e = c * CHUNK + el;
            if (e < E) {
                // 16-lane broadcast reads from LDS (ds_load_b64, no conflicts)
                const long long r  = (long long)bb[el * 2 + 0];
                const long long ci = (long long)bb[el * 2 + 1];
                const float4 v = *(const float4*)(x + ci * HIDDEN + l4 * 4);
                float* o = out + r * HIDDEN + l4 * 4;
                hw_atomic_add_f32(o + 0, v.x);
                hw_atomic_add_f32(o + 1, v.y);
                hw_atomic_add_f32(o + 2, v.z);
                hw_atomic_add_f32(o + 3, v.w);
            }
        }
        buf ^= 1;
    }
}

// ---------------------------------------------------------------------------
extern "C" void kernel_launch(void* const* d_in, const int* in_sizes, int n_in,
                              void* d_out, int out_size, void* d_ws, size_t ws_size,
                              hipStream_t stream) {
    const float*      x    = (const float*)d_in[0];
    const long long*  ei   = (const long long*)d_in[1];   // [2, E] int64
    const float*      eps  = (const float*)d_in[2];
    float*            out  = (float*)d_out;

    const long long E = (long long)(in_sizes[1] / 2);
    const int total4  = in_sizes[0] / 4;                  // N * HIDDEN / 4

    const long long* rows = ei;        // edge_index[0] = destinations
    const long long* cols = ei + E;    // edge_index[1] = sources

    // 1) out = (1 + eps) * x
    int gridC = (total4 + THREADS - 1) / THREADS;
    gin_combine<<<gridC, THREADS, 0, stream>>>(x, eps, out, total4);

    // 2) out[row] += x[col]   (L2-resident scatter atomics)
    const int blocks = 1024;
    const int totalWaves = blocks * WAVES_PER_BLOCK;
    gin_scatter<<<blocks, THREADS, 0, stream>>>(x, rows, cols, out, E, totalWaves);
}